// SpatialTransformer_9835475108625
// MI455X (gfx1250) — compile-verified
//
#include <hip/hip_runtime.h>

// Spatial transformer bilinear sampling, MI455X (gfx1250, wave32).
//
// - Affine coordinate transform done with V_WMMA_F32_16X16X4_F32:
//     A (16x4): rows 0,8 = (t0,t1,t2,0), rows 1,9 = (t3,t4,t5,0)
//     B (4x16): column n = (gx_n, gy_n, 1, 0)
//   D layout gives lane n and lane 16+n the same pixel's (x,y) in d[0],d[1],
//   so the two lanes of each pixel split the 16 channels (8 each).
// - Input (134MB) fits in the 192MB L2; output stores are non-temporal so the
//   streaming writes don't evict it -> gathers hit L2.

typedef __attribute__((ext_vector_type(2))) float v2f;
typedef __attribute__((ext_vector_type(8))) float v8f;

constexpr int Bn = 32, Cn = 16, Hn = 256, Wn = 256;
constexpr int HW = Hn * Wn;                 // 65536
constexpr int PIX_PER_WAVE = 16;

__global__ __launch_bounds__(256) void stn_bilinear_kernel(
    const float* __restrict__ in, const float* __restrict__ theta,
    float* __restrict__ out)
{
    const int tid  = blockIdx.x * blockDim.x + threadIdx.x;
    const int lane = threadIdx.x & 31;
    const int wave = tid >> 5;
    const int n    = lane & 15;   // matrix column = pixel index within wave
    const int hi   = lane >> 4;   // 0: K=0,1 half / channels 0-7 ; 1: K=2,3 half / channels 8-15

    const int pixBase = wave * PIX_PER_WAVE;      // first pixel of this wave
    const int b   = pixBase >> 16;                // / (H*W)
    const int rem = pixBase & (HW - 1);
    const int oy  = rem >> 8;                     // / W
    const int ox  = (rem & (Wn - 1)) + n;

    // batch index is wave-uniform -> force scalar theta loads
    const int bu = __builtin_amdgcn_readfirstlane(b);
    const float t0 = theta[bu * 6 + 0], t1 = theta[bu * 6 + 1], t2 = theta[bu * 6 + 2];
    const float t3 = theta[bu * 6 + 3], t4 = theta[bu * 6 + 4], t5 = theta[bu * 6 + 5];

    // ---- A matrix 16x4 f32. Lane layout: lanes 0-15 hold A[m][0..1] (m=lane),
    //      lanes 16-31 hold A[m][2..3] (m=lane-16).
    const bool rowx = (n == 0) | (n == 8);
    const bool rowy = (n == 1) | (n == 9);
    v2f a;
    a.x = rowx ? (hi ? t2 : t0) : (rowy ? (hi ? t5 : t3) : 0.0f);
    a.y = rowx ? (hi ? 0.0f : t1) : (rowy ? (hi ? 0.0f : t4) : 0.0f);

    // ---- B matrix 4x16 f32. Lanes 0-15 hold B[0..1][n], lanes 16-31 hold B[2..3][n].
    const float gx = -1.0f + (float)ox * (2.0f / (Wn - 1));
    const float gy = -1.0f + (float)oy * (2.0f / (Hn - 1));
    v2f bm;
    bm.x = hi ? 1.0f : gx;
    bm.y = hi ? 0.0f : gy;

    v8f acc = {};
    // D = A x B + 0 :
    //   lane n    : acc[0] = x_n (row 0), acc[1] = y_n (row 1)
    //   lane 16+n : acc[0] = x_n (row 8), acc[1] = y_n (row 9)
    acc = __builtin_amdgcn_wmma_f32_16x16x4_f32(
        /*neg_a=*/false, a, /*neg_b=*/false, bm,
        /*c_mod=*/(short)0, acc, /*reuse_a=*/false, /*reuse_b=*/false);

    // ---- bilinear sampling (matches reference semantics exactly)
    float fx = fminf(fmaxf(acc[0], -1.0f), 1.0f);
    float fy = fminf(fmaxf(acc[1], -1.0f), 1.0f);
    fx = (fx + 1.0f) * (0.5f * (Wn - 1));
    fy = (fy + 1.0f) * (0.5f * (Hn - 1));
    const float x0f = floorf(fx), y0f = floorf(fy);
    const int x0 = (int)x0f, y0 = (int)y0f;
    const int x1 = min(x0 + 1, Wn - 1);
    const int y1 = min(y0 + 1, Hn - 1);
    const float lx = fx - x0f, ly = fy - y0f;
    const float wa = (1.0f - lx) * (1.0f - ly);
    const float wb = (1.0f - lx) * ly;
    const float wc = lx * (1.0f - ly);
    const float wd = lx * ly;

    const int o00 = y0 * Wn + x0;
    const int o10 = y1 * Wn + x0;
    const int o01 = y0 * Wn + x1;
    const int o11 = y1 * Wn + x1;

    const int c0 = hi * 8;
    const float* p = in  + (size_t)(b * Cn + c0) * HW;
    float*       q = out + (size_t)(b * Cn + c0) * HW + oy * Wn + ox;

#pragma unroll
    for (int i = 0; i < 8; ++i) {
        const float Ia = p[o00 + i * HW];
        const float Ib = p[o10 + i * HW];
        const float Ic = p[o01 + i * HW];
        const float Id = p[o11 + i * HW];
        const float r  = wa * Ia + wb * Ib + wc * Ic + wd * Id;
        __builtin_nontemporal_store(r, q + i * HW);   // NT: don't evict input from L2
    }
}

extern "C" void kernel_launch(void* const* d_in, const int* in_sizes, int n_in,
                              void* d_out, int out_size, void* d_ws, size_t ws_size,
                              hipStream_t stream) {
    const float* in    = (const float*)d_in[0];   // [32,16,256,256] f32
    const float* theta = (const float*)d_in[1];   // [32,6] f32
    float* out = (float*)d_out;                   // [32,16,256,256] f32

    const int totalPix  = Bn * Hn * Wn;                 // 2,097,152
    const int waves     = totalPix / PIX_PER_WAVE;      // 131,072
    const int blockSize = 256;                          // 8 waves / block
    const int blocks    = (waves * 32) / blockSize;     // 16,384

    hipLaunchKernelGGL(stn_bilinear_kernel, dim3(blocks), dim3(blockSize), 0, stream,
                       in, theta, out);
}